// SoftMLoss_8392366096764
// MI455X (gfx1250) — compile-verified
//
#include <hip/hip_runtime.h>
#include <math.h>

typedef float v2f __attribute__((ext_vector_type(2)));
typedef float v8f __attribute__((ext_vector_type(8)));
typedef int   v4i __attribute__((ext_vector_type(4)));

constexpr int BDIM  = 512;            // batch
constexpr int DDIM  = 32768;          // feature dim
constexpr int FRP   = 8;              // frames per camera block
constexpr int NROWS = 1792;           // 64 blocks * C(8,2)
constexpr int LLEN  = BDIM - FRP + 1; // 505

constexpr int TC  = 64;               // output tile (TC x TC per block)
constexpr int KC  = 32;               // K chunk staged in LDS
constexpr int LDW = 36;               // padded LDS row stride (dwords): 144B, 16B-aligned,
                                      // 36*r mod 64 distinct for r=0..15 -> conflict-free
constexpr int SPLITK = 4;             // K split across blockIdx.z
constexpr int KSLICE = DDIM / SPLITK; // 8192
constexpr int NCHUNK = KSLICE / KC;   // 256

// ---- gfx1250 async global->LDS copy (fall back to load+ds_store) ----------
#if defined(__has_builtin)
#if __has_builtin(__builtin_amdgcn_global_load_async_to_lds_b128)
#define HAVE_ASYNC_LDS 1
#endif
#endif

#if defined(HAVE_ASYNC_LDS)
// signature (from clang diagnostic): (AS1 int4*, AS3 int4*, imm offset, imm cpol)
#define ASYNC_COPY16(gp, lp, off)                                              \
  __builtin_amdgcn_global_load_async_to_lds_b128(                              \
      (__attribute__((address_space(1))) v4i*)(gp),                            \
      (__attribute__((address_space(3))) v4i*)(lp), (off), 0)
#if __has_builtin(__builtin_amdgcn_s_wait_asynccnt)
#define WAIT_ASYNC() __builtin_amdgcn_s_wait_asynccnt(0)
#else
#define WAIT_ASYNC() asm volatile("s_wait_asynccnt 0x0" ::: "memory")
#endif
#else
#define ASYNC_COPY16(gp, lp, off)                                              \
  do { *(float4*)((char*)(lp) + (off)) =                                       \
         *(const float4*)((const char*)(gp) + (off)); } while (0)
#define WAIT_ASYNC() do {} while (0)
#endif

// ---------------------------------------------------------------------------
// Kernel 1: partial gram slices via V_WMMA_F32_16X16X4_F32.
// Grid (8,8,4): 64x64 output tile per block, K split 4 ways (blockIdx.z).
// 8 waves: wave w -> M-subtile (w&3), N-subtiles 2*(w>>2), 2*(w>>2)+1.
// Double-buffered LDS staging of A(64xKC) and B(64xKC) with async copies.
//
// FP32 WMMA fragment layout (ISA 7.12.2): lane<16 holds K=0,1 / lane>=16
// holds K=2,3 for row (lane&15) -> one 8B LDS read per fragment.
// ---------------------------------------------------------------------------
__global__ __launch_bounds__(256) void gram_wmma(const float* __restrict__ x,
                                                 float* __restrict__ part) {
  const int bm   = blockIdx.x;
  const int bn   = blockIdx.y;
  const int ks   = blockIdx.z;
  const int tid  = threadIdx.x;
  const int wave = tid >> 5;
  const int lane = tid & 31;

  __shared__ float As[2][TC * LDW];
  __shared__ float Bs[2][TC * LDW];

  // --- staging addressing: thread t copies row t>>2, 8-dword segment t&3 ---
  const int srow = tid >> 2;
  const int sseg = (tid & 3) << 3;   // dword offset 0,8,16,24
  const int kbase0 = ks * KSLICE;
  const float* ga = x + (size_t)(bm * TC + srow) * DDIM + kbase0 + sseg;
  const float* gb = x + (size_t)(bn * TC + srow) * DDIM + kbase0 + sseg;
  const int lofs = srow * LDW + sseg;

  // --- fragment addressing ---
  const int wm  = wave & 3;
  const int wn2 = (wave >> 2) << 1;  // first of two N subtiles
  const int fr  = lane & 15;
  const int fk  = (lane >> 4) << 1;  // 0 or 2
  const int aofs  = (wm * 16 + fr) * LDW + fk;
  const int b0ofs = (wn2 * 16 + fr) * LDW + fk;
  const int b1ofs = ((wn2 + 1) * 16 + fr) * LDW + fk;

  v8f acc0 = {};
  v8f acc1 = {};

  // prologue: stage chunk 0 into buffer 0
  {
    ASYNC_COPY16(ga, &As[0][lofs], 0);
    ASYNC_COPY16(ga, &As[0][lofs], 16);
    ASYNC_COPY16(gb, &Bs[0][lofs], 0);
    ASYNC_COPY16(gb, &Bs[0][lofs], 16);
    WAIT_ASYNC();
    __syncthreads();
  }

  for (int c = 0; c < NCHUNK; ++c) {
    const int buf = c & 1;
    if (c + 1 < NCHUNK) {
      const int kd = (c + 1) * KC;   // dword advance into this K slice
      ASYNC_COPY16(ga + kd, &As[buf ^ 1][lofs], 0);
      ASYNC_COPY16(ga + kd, &As[buf ^ 1][lofs], 16);
      ASYNC_COPY16(gb + kd, &Bs[buf ^ 1][lofs], 0);
      ASYNC_COPY16(gb + kd, &Bs[buf ^ 1][lofs], 16);
    }

    const float* aF  = &As[buf][aofs];
    const float* bF0 = &Bs[buf][b0ofs];
    const float* bF1 = &Bs[buf][b1ofs];
#pragma unroll
    for (int s = 0; s < KC / 4; ++s) {
      v2f a  = *(const v2f*)(aF  + s * 4);
      v2f b0 = *(const v2f*)(bF0 + s * 4);
      v2f b1 = *(const v2f*)(bF1 + s * 4);
      acc0 = __builtin_amdgcn_wmma_f32_16x16x4_f32(
          false, a, false, b0, (short)0, acc0, false, false);
      acc1 = __builtin_amdgcn_wmma_f32_16x16x4_f32(
          false, a, false, b1, (short)0, acc1, false, false);
    }

    WAIT_ASYNC();
    __syncthreads();
  }

  // --- epilogue: C/D layout (vgpr i, lane l) -> M=i+8*(l>=16), N=l&15 ---
  float* outp = part + (size_t)ks * BDIM * BDIM;
  const int Mb = bm * TC + wm * 16 + ((lane >> 4) << 3);
  const int N0 = bn * TC + wn2 * 16 + (lane & 15);
#pragma unroll
  for (int i = 0; i < 8; ++i) {
    outp[(size_t)(Mb + i) * BDIM + N0]      = acc0[i];
    outp[(size_t)(Mb + i) * BDIM + N0 + 16] = acc1[i];
  }
}

// ---------------------------------------------------------------------------
// Kernel 2: sum the SPLITK partial slices into slice 0 (= gram).
// ---------------------------------------------------------------------------
__global__ __launch_bounds__(256) void reduce_partials(float* __restrict__ part) {
  const size_t i = (size_t)blockIdx.x * 256 + threadIdx.x;
  constexpr size_t N = (size_t)BDIM * BDIM;
  float s = part[i];
#pragma unroll
  for (int k = 1; k < SPLITK; ++k) s += part[i + (size_t)k * N];
  part[i] = s;
}

// ---------------------------------------------------------------------------
// Kernel 3: one block per logit row (gram diagonal == squared norms).
// ---------------------------------------------------------------------------
__global__ __launch_bounds__(128) void loss_rows(const float* __restrict__ gram,
                                                 float* __restrict__ rowvals) {
  const int t = blockIdx.x;
  const int blkI = t / 28;
  int p = t % 28;
  int a = 0;
  while (p >= 7 - a) { p -= 7 - a; ++a; }   // unrank pair (a,b), 0<=a<b<8
  const int b  = a + 1 + p;
  const int r  = blkI * FRP + a;
  const int ii = blkI * FRP + b;
  const int cut = blkI * FRP;               // label position

  const float grr = gram[(size_t)r * BDIM + r];

  float lv[4];
  int   cnt  = 0;
  float lmin = 3.4e38f;
  for (int j = threadIdx.x; j < LLEN; j += 128) {
    const int c = (j < cut) ? j : ((j == cut) ? ii : j + FRP - 1);
    const float d2 = grr + gram[(size_t)c * BDIM + c]
                   - 2.0f * gram[(size_t)r * BDIM + c];
    const float lg = sqrtf(fmaxf(d2, 0.0f));
    lv[cnt++] = lg;
    lmin = fminf(lmin, lg);
  }

  __shared__ float sred[128];
  sred[threadIdx.x] = lmin;
  __syncthreads();
  for (int s = 64; s > 0; s >>= 1) {
    if (threadIdx.x < s) sred[threadIdx.x] = fminf(sred[threadIdx.x], sred[threadIdx.x + s]);
    __syncthreads();
  }
  lmin = sred[0];
  __syncthreads();

  float se = 0.0f;
  for (int q = 0; q < cnt; ++q) se += expf(lmin - lv[q]);
  sred[threadIdx.x] = se;
  __syncthreads();
  for (int s = 64; s > 0; s >>= 1) {
    if (threadIdx.x < s) sred[threadIdx.x] += sred[threadIdx.x + s];
    __syncthreads();
  }

  if (threadIdx.x == 0) {
    const float pos = sqrtf(fmaxf(
        grr + gram[(size_t)ii * BDIM + ii] - 2.0f * gram[(size_t)r * BDIM + ii], 0.0f));
    rowvals[t] = pos + (-lmin + logf(sred[0]));  // logsumexp(-l)
  }
}

// ---------------------------------------------------------------------------
// Kernel 4: mean of the 1792 row values -> scalar loss.
// ---------------------------------------------------------------------------
__global__ __launch_bounds__(256) void final_reduce(const float* __restrict__ rowvals,
                                                    float* __restrict__ out) {
  __shared__ float sred[256];
  float s = 0.0f;
  for (int i = threadIdx.x; i < NROWS; i += 256) s += rowvals[i];
  sred[threadIdx.x] = s;
  __syncthreads();
  for (int st = 128; st > 0; st >>= 1) {
    if (threadIdx.x < st) sred[threadIdx.x] += sred[threadIdx.x + st];
    __syncthreads();
  }
  if (threadIdx.x == 0) out[0] = sred[0] / (float)NROWS;
}

extern "C" void kernel_launch(void* const* d_in, const int* in_sizes, int n_in,
                              void* d_out, int out_size, void* d_ws, size_t ws_size,
                              hipStream_t stream) {
  (void)in_sizes; (void)n_in; (void)out_size; (void)ws_size;
  const float* x = (const float*)d_in[0];

  // workspace: SPLITK gram slices (4 MB), slice 0 becomes gram; then rowvals
  float* part    = (float*)d_ws;
  float* rowvals = (float*)((char*)d_ws +
                            (size_t)SPLITK * BDIM * BDIM * sizeof(float));

  dim3 ggrid(BDIM / TC, BDIM / TC, SPLITK);            // 8 x 8 x 4
  gram_wmma<<<ggrid, 256, 0, stream>>>(x, part);
  reduce_partials<<<(BDIM * BDIM) / 256, 256, 0, stream>>>(part);
  loss_rows<<<NROWS, 128, 0, stream>>>(part, rowvals); // gram == slice 0
  final_reduce<<<1, 256, 0, stream>>>(rowvals, (float*)d_out);
}